// HarmonicGenerator_19301583029002
// MI455X (gfx1250) — compile-verified
//
#include <hip/hip_runtime.h>
#include <hip/hip_bf16.h>
#include <math.h>

#define N_HARM   80
#define T_FRAMES 500
#define BATCH    4
#define HOP      240
#define SR       24000
#define NS       120240                 /* (T+1)*HOP */
#define TWO_PI_F 6.28318530717958647692f
#define SPW      16                     /* samples per wave */
#define WAVES_TOTAL (BATCH * (NS / SPW))

typedef __attribute__((ext_vector_type(2))) float v2f;
typedef __attribute__((ext_vector_type(8))) float v8f;

// ---------------------------------------------------------------------------
// Kernel 1: per-(b,h) sequential phase accumulation (matches jnp.cumsum order
// in f32) + mod 2*pi. 320 independent chains of length 500 -> trivial.
// phases laid out (B, T, N_HARM) for broadcast-friendly reads in kernel 2.
// ---------------------------------------------------------------------------
__global__ void phase_cumsum_kernel(const float* __restrict__ f0,
                                    const float* __restrict__ init_phase,
                                    float* __restrict__ phases) {
    int tid = blockIdx.x * blockDim.x + threadIdx.x;
    if (tid >= BATCH * N_HARM) return;
    int b = tid / N_HARM;
    int h = tid % N_HARM;
    float hf  = (float)(h + 1);
    float acc = init_phase[b * N_HARM + h];
    const float hopsr = (float)HOP / (float)SR;     // 0.01f, exact
    for (int t = 0; t < T_FRAMES; ++t) {
        // exclusive prefix: store BEFORE adding this frame's increment
        phases[((size_t)b * T_FRAMES + t) * N_HARM + h] = fmodf(acc, TWO_PI_F);
        float inc = (TWO_PI_F * (f0[b * T_FRAMES + t] * hf)) * hopsr;
        acc += inc;
    }
}

// ---------------------------------------------------------------------------
// ADSR envelope for one sample, replicating the reference's 8-note
// overwrite-in-order jnp.where semantics.
// ---------------------------------------------------------------------------
__device__ __forceinline__ float adsr_env(int s, int b,
                                          const float* __restrict__ adsr,
                                          const int* __restrict__ on_f,
                                          const int* __restrict__ off_f) {
    float env = 0.0f;
    float sf  = (float)s;
#pragma unroll
    for (int n = 0; n < 8; ++n) {
        int on  = on_f [b * 8 + n];
        int off = off_f[b * 8 + n];
        int ns  = on * HOP;
        int ne  = min(off * HOP, NS);
        bool valid = (ns >= 0) && (ns < NS);
        int onset = min(on, T_FRAMES - 1);          // jax clamp-indexing
        const float* p = &adsr[((size_t)b * T_FRAMES + onset) * 4];
        int   a_n = (int)floorf(p[0] * (float)SR);
        int   d_n = (int)floorf(p[1] * (float)SR);
        float sus = p[2];
        int   r_n = (int)floorf(p[3] * (float)SR);
        int a_end = min(ns + a_n, NS);
        int d_end = min(a_end + d_n, NS);
        int r_end = min(ne + r_n, NS);
        float att = (sf - (float)ns) / (float)max(a_n, 1);
        float dec = 1.0f - (1.0f - sus) * (sf - (float)a_end) / (float)max(d_n, 1);
        float rel = sus * (1.0f - (sf - (float)ne) / (float)max(r_n, 1));
        if (valid && s >= ns    && s < a_end) env = att;
        if (valid && s >= a_end && s < d_end) env = dec;
        if (valid && s >= d_end && s < ne   ) env = sus;
        if (valid && s >= ne    && s < r_end) env = rel;
    }
    return env;
}

// ---------------------------------------------------------------------------
// Kernel 2: one wave32 per 16 consecutive samples of one batch row.
// Harmonic reduction via V_WMMA_F32_16X16X4_F32 with B = ones:
//   A (16x4 f32): lanes 0-15 hold sample m, harmonics {k,k+1} in VGPR0/1;
//                 lanes 16-31 hold sample m, harmonics {k+2,k+3}.
//   D = A x Ones + C -> every column of row m holds sum over this K-chunk.
// 20 chained WMMAs cover all 80 harmonics at full f32 precision.
// ---------------------------------------------------------------------------
__global__ void __launch_bounds__(256) synth_kernel(
        const float* __restrict__ f0,      // (B,T)
        const float* __restrict__ amps,    // (B,T,80)
        const float* __restrict__ vmix,    // (B,T,1)
        const float* __restrict__ adsr,    // (B,T,4)
        const int*   __restrict__ on_f,    // (B,8)
        const int*   __restrict__ off_f,   // (B,8)
        const float* __restrict__ phases,  // (B,T,80) workspace
        float*       __restrict__ out)     // (B,NS)
{
    int lane = threadIdx.x & 31;
    int wave = blockIdx.x * (blockDim.x >> 5) + (threadIdx.x >> 5);
    if (wave >= WAVES_TOTAL) return;       // wave-uniform: EXEC stays all-1s

    const int wavesPerB = NS / SPW;
    int b  = wave / wavesPerB;
    int s0 = (wave - b * wavesPerB) * SPW;

    // This lane's product sample (lanes L and L+16 share sample s0+(L&15))
    int   m   = lane & 15;
    int   s   = s0 + m;
    int   idx = min(s / HOP, T_FRAMES - 1);
    float t   = (float)s / (float)SR;
    float f0v = f0[b * T_FRAMES + idx];

    // linear-interp weights for harmonic amplitudes at sample s
    const float scale = (float)T_FRAMES / (float)NS;
    float coord = fmaxf(((float)s + 0.5f) * scale - 0.5f, 0.0f);
    int   i0 = (int)floorf(coord);
    int   i1 = min(i0 + 1, T_FRAMES - 1);
    float w  = coord - (float)i0;
    const float* amp0 = &amps  [((size_t)b * T_FRAMES + i0 ) * N_HARM];
    const float* amp1 = &amps  [((size_t)b * T_FRAMES + i1 ) * N_HARM];
    const float* ph   = &phases[((size_t)b * T_FRAMES + idx) * N_HARM];

    int hoff = (lane < 16) ? 0 : 2;        // K-offset per ISA A-matrix layout

    v8f acc = {};
    v2f ones; ones.x = 1.0f; ones.y = 1.0f;

    for (int c = 0; c < N_HARM / 4; ++c) {
        v2f a;
#pragma unroll
        for (int j = 0; j < 2; ++j) {
            int   hm1 = 4 * c + hoff + j;              // 0-indexed harmonic
            float hf  = (float)(hm1 + 1);
            float arg = TWO_PI_F * (f0v * hf) * t + ph[hm1];
            float sv  = sinf(arg);
            float av  = amp0[hm1] * (1.0f - w) + amp1[hm1] * w;
            float pr  = sv * av;
            if (j == 0) a.x = pr; else a.y = pr;
        }
        acc = __builtin_amdgcn_wmma_f32_16x16x4_f32(
                  false, a, false, ones, (short)0, acc, false, false);
    }

    // C/D layout: lane<16 has rows 0-7 in acc[0..7]; lane>=16 has rows 8-15.
    // Writers: lanes 0-7 emit rows 0-7, lanes 16-23 emit rows 8-15.
    int sub = lane & 15;
    if (sub < 8) {
        float val = 0.0f;
#pragma unroll
        for (int j = 0; j < 8; ++j) if (j == sub) val = acc[j];
        int so = s0 + ((lane < 16) ? sub : (8 + sub));
        float env = adsr_env(so, b, adsr, on_f, off_f);
        float c2 = fmaxf(((float)so + 0.5f) * scale - 0.5f, 0.0f);
        int   j0 = (int)floorf(c2);
        int   j1 = min(j0 + 1, T_FRAMES - 1);
        float w2 = c2 - (float)j0;
        float vm = vmix[b * T_FRAMES + j0] * (1.0f - w2)
                 + vmix[b * T_FRAMES + j1] * w2;
        out[(size_t)b * NS + so] = val * env * vm;
    }
}

extern "C" void kernel_launch(void* const* d_in, const int* in_sizes, int n_in,
                              void* d_out, int out_size, void* d_ws, size_t ws_size,
                              hipStream_t stream) {
    (void)in_sizes; (void)n_in; (void)out_size; (void)ws_size;
    const float* f0         = (const float*)d_in[0];
    const float* amps       = (const float*)d_in[1];
    const float* vmix       = (const float*)d_in[2];
    const float* adsr       = (const float*)d_in[3];
    const float* init_phase = (const float*)d_in[4];
    const int*   on_f       = (const int*)  d_in[5];
    const int*   off_f      = (const int*)  d_in[6];
    // d_in[7] = duration scalar; fixed to (T+1)*HOP = 120240 per setup.
    float* phases = (float*)d_ws;          // B*T*N_HARM floats = 640 KB
    float* out    = (float*)d_out;

    phase_cumsum_kernel<<<(BATCH * N_HARM + 63) / 64, 64, 0, stream>>>(
        f0, init_phase, phases);

    const int wavesPerBlock = 256 / 32;
    int blocks = (WAVES_TOTAL + wavesPerBlock - 1) / wavesPerBlock;
    synth_kernel<<<blocks, 256, 0, stream>>>(
        f0, amps, vmix, adsr, on_f, off_f, phases, out);
}